// GIN_87497073754464
// MI455X (gfx1250) — compile-verified
//
#include <hip/hip_runtime.h>
#include <hip/hip_bf16.h>

// ---------------------------------------------------------------------------
// GINE (3x GINEConv + mol-pool + linear head) for MI455X (gfx1250, wave32).
// All GEMMs use V_WMMA_F32_16X16X4_F32 (full fp32 precision on the matrix
// pipe). Gathers/scatters exploit the 192MB L2 (node features are ~102MB).
// All global offsets are 32-bit (max ~102MB) so the compiler can use the
// SGPR-base + 32-bit-VGPR-offset (GVS) addressing form.
// ---------------------------------------------------------------------------

typedef float v2f __attribute__((ext_vector_type(2)));
typedef float v8f __attribute__((ext_vector_type(8)));

#define N_NODES   200000
#define N_EDGES   1000000
#define IN_DIM    64
#define EDGE_DIM  16
#define HDIM      128
#define N_MOLS    8000
#define BN_EPS    1e-5f
#define SLOPE     0.01f

// ---------------------------------------------------------------------------
__global__ void zero_kernel(float* __restrict__ p, unsigned n) {
  unsigned i = blockIdx.x * blockDim.x + threadIdx.x;
  unsigned stride = gridDim.x * blockDim.x;
  for (; i < n; i += stride) p[i] = 0.f;
}

// ---------------------------------------------------------------------------
// Fused edge-message kernel:
//   m[e] = relu( x[src[e]] + edge_attr[e] @ We^T + be )
//   agg[dst[e]] += m[e]     (atomic scatter-add)
// One wave per 16-edge tile. Edge linear done with f32 WMMA:
//   A = edge_attr tile [16 x 16], B = We^T [16 x D], 4 k-steps of 16x16x4.
template<int D>
__global__ void edge_msg_kernel(const float* __restrict__ xin,       // [N,D]
                                const int*   __restrict__ src,
                                const int*   __restrict__ dst,
                                const float* __restrict__ edge_attr, // [E,16]
                                const float* __restrict__ We,        // [D,16]
                                const float* __restrict__ be,        // [D]
                                float*       __restrict__ agg)       // [N,D]
{
  const unsigned wave = threadIdx.x >> 5;
  const unsigned lane = threadIdx.x & 31;
  const unsigned row  = lane & 15;          // M (A) / N (B,C) coordinate
  const unsigned koff = (lane >> 4) << 1;   // 0 for lanes 0-15, 2 for 16-31
  const unsigned tile = blockIdx.x * (blockDim.x >> 5) + wave;
  if (tile >= N_EDGES / 16) return;    // wave-uniform exit (EXEC stays all-1s)
  const unsigned e0 = tile * 16;

  // A fragments for all 4 k-steps (ISA 7.12.2: 32-bit A 16x4 layout)
  v2f a[4];
  const unsigned abase = (e0 + row) * EDGE_DIM;
  #pragma unroll
  for (int ks = 0; ks < 4; ++ks) {
    a[ks].x = edge_attr[abase + 4 * ks + koff + 0];
    a[ks].y = edge_attr[abase + 4 * ks + koff + 1];
  }

  // Edge indices for the 8 C-fragment rows owned by this lane half.
  const unsigned mbase = (lane >> 4) ? 8u : 0u;
  unsigned soff[8], doff[8];             // 32-bit element offsets (row base)
  #pragma unroll
  for (int j = 0; j < 8; ++j) {
    const unsigned e = e0 + mbase + j;
    soff[j] = (unsigned)src[e] * (unsigned)D;
    doff[j] = (unsigned)dst[e] * (unsigned)D;
  }

  #pragma unroll
  for (int nt = 0; nt < D / 16; ++nt) {
    const unsigned ch = nt * 16 + row;   // output channel for this lane
    v8f c = {0.f, 0.f, 0.f, 0.f, 0.f, 0.f, 0.f, 0.f};
    const unsigned wbase = ch * EDGE_DIM;
    #pragma unroll
    for (int ks = 0; ks < 4; ++ks) {
      v2f b;
      b.x = We[wbase + 4 * ks + koff + 0];
      b.y = We[wbase + 4 * ks + koff + 1];
      c = __builtin_amdgcn_wmma_f32_16x16x4_f32(false, a[ks], false, b,
                                                (short)0, c, false, false);
    }
    const float bias = be[ch];
    #pragma unroll
    for (int j = 0; j < 8; ++j) {
      // C layout: VGPR j -> edge row (mbase+j), lane%16 -> channel.
      // Lanes 0-15 hit 16 consecutive channels -> coalesced 64B gather/scatter.
      float v = c[j] + bias + xin[soff[j] + ch];
      v = fmaxf(v, 0.f);
      atomicAdd(&agg[doff[j] + ch], v);
    }
  }
}

// ---------------------------------------------------------------------------
// Node update: h = leakyrelu( BN( (x + agg) @ W^T + b ) )
// One wave per 16-node tile, 8 accumulator fragments (full 128 out channels),
// K-loop of f32 WMMA.
template<int D>
__global__ void node_gemm_kernel(const float* __restrict__ xin,   // [N,D]
                                 const float* __restrict__ agg,   // [N,D]
                                 const float* __restrict__ W,     // [128,D]
                                 const float* __restrict__ bias,  // [128]
                                 const float* __restrict__ gamma,
                                 const float* __restrict__ beta,
                                 const float* __restrict__ rmean,
                                 const float* __restrict__ rvar,
                                 float*       __restrict__ hout)  // [N,128]
{
  const unsigned wave = threadIdx.x >> 5;
  const unsigned lane = threadIdx.x & 31;
  const unsigned row  = lane & 15;
  const unsigned koff = (lane >> 4) << 1;
  const unsigned tile = blockIdx.x * (blockDim.x >> 5) + wave;
  if (tile >= N_NODES / 16) return;    // wave-uniform
  const unsigned m0 = tile * 16;
  const unsigned arow = (m0 + row) * (unsigned)D;

  v8f acc[HDIM / 16];
  const v8f vzero = {0.f, 0.f, 0.f, 0.f, 0.f, 0.f, 0.f, 0.f};
  #pragma unroll
  for (int nt = 0; nt < HDIM / 16; ++nt) acc[nt] = vzero;

  for (int ks = 0; ks < D / 4; ++ks) {
    const unsigned k = 4 * ks + koff;
    v2f a;
    a.x = xin[arow + k]     + agg[arow + k];
    a.y = xin[arow + k + 1] + agg[arow + k + 1];
    #pragma unroll
    for (int nt = 0; nt < HDIM / 16; ++nt) {
      const unsigned wrow = (nt * 16 + row) * (unsigned)D;  // 64KB, L2-hot
      v2f b;
      b.x = W[wrow + k];
      b.y = W[wrow + k + 1];
      acc[nt] = __builtin_amdgcn_wmma_f32_16x16x4_f32(false, a, false, b,
                                                      (short)0, acc[nt],
                                                      false, false);
    }
  }

  const unsigned mbase = (lane >> 4) ? 8u : 0u;
  #pragma unroll
  for (int nt = 0; nt < HDIM / 16; ++nt) {
    const unsigned ch = nt * 16 + row;
    const float inv = rsqrtf(rvar[ch] + BN_EPS);
    const float g = gamma[ch], bt = beta[ch], mu = rmean[ch], bb = bias[ch];
    #pragma unroll
    for (int j = 0; j < 8; ++j) {
      const unsigned node = m0 + mbase + j;
      float h = acc[nt][j] + bb;
      h = (h - mu) * inv * g + bt;
      h = (h > 0.f) ? h : SLOPE * h;
      hout[node * HDIM + ch] = h;       // coalesced 64B per half-wave
    }
  }
}

// ---------------------------------------------------------------------------
// partial[n] (+)= dot(h[n, 0:128], wl[0:128])  -- per-layer head contribution
__global__ void dot_accum_kernel(const float* __restrict__ h,
                                 const float* __restrict__ wl,
                                 float* __restrict__ partial,
                                 int accumulate)
{
  const unsigned n = blockIdx.x * blockDim.x + threadIdx.x;
  if (n >= N_NODES) return;
  const float4* hr = (const float4*)(h + n * HDIM);
  const float4* w4 = (const float4*)wl;
  float s = 0.f;
  #pragma unroll 8
  for (int c = 0; c < HDIM / 4; ++c) {
    const float4 hv = hr[c];
    const float4 wv = w4[c];
    s += hv.x * wv.x + hv.y * wv.y + hv.z * wv.z + hv.w * wv.w;
  }
  partial[n] = accumulate ? (partial[n] + s) : s;
}

// ---------------------------------------------------------------------------
// hpool[batch[n]] += h3[n]   (molecule pooling; hpool is 4MB -> L2 resident)
__global__ void pool_kernel(const float* __restrict__ h3,
                            const int*   __restrict__ batch,
                            float*       __restrict__ hpool)
{
  const unsigned i = blockIdx.x * blockDim.x + threadIdx.x;
  if (i >= (unsigned)N_NODES * HDIM) return;
  const unsigned n  = i >> 7;
  const unsigned ch = i & 127;
  atomicAdd(&hpool[(unsigned)batch[n] * HDIM + ch], h3[i]);
}

// ---------------------------------------------------------------------------
// out[n] = sigmoid( partial[n] + dot(hpool[batch[n]], wl4) + bl )
__global__ void final_kernel(const float* __restrict__ partial,
                             const int*   __restrict__ batch,
                             const float* __restrict__ hpool,
                             const float* __restrict__ wl4,
                             const float* __restrict__ bl,
                             float*       __restrict__ out)
{
  const unsigned n = blockIdx.x * blockDim.x + threadIdx.x;
  if (n >= N_NODES) return;
  const float4* hp = (const float4*)(hpool + (unsigned)batch[n] * HDIM);
  const float4* w4 = (const float4*)wl4;
  float s = partial[n] + bl[0];
  #pragma unroll 8
  for (int c = 0; c < HDIM / 4; ++c) {
    const float4 hv = hp[c];
    const float4 wv = w4[c];
    s += hv.x * wv.x + hv.y * wv.y + hv.z * wv.z + hv.w * wv.w;
  }
  out[n] = 1.f / (1.f + expf(-s));
}

// ---------------------------------------------------------------------------
extern "C" void kernel_launch(void* const* d_in, const int* in_sizes, int n_in,
                              void* d_out, int out_size, void* d_ws, size_t ws_size,
                              hipStream_t stream) {
  const float* x          = (const float*)d_in[0];
  const int*   edge_index = (const int*)  d_in[1];
  const float* edge_attr  = (const float*)d_in[2];
  const int*   batch      = (const int*)  d_in[3];

  const float *We[3], *be[3], *W[3], *b[3], *gam[3], *bet[3], *rm[3], *rv[3];
  for (int i = 0; i < 3; ++i) {
    const int base = 4 + i * 8;
    We[i]  = (const float*)d_in[base + 0];
    be[i]  = (const float*)d_in[base + 1];
    W[i]   = (const float*)d_in[base + 2];
    b[i]   = (const float*)d_in[base + 3];
    gam[i] = (const float*)d_in[base + 4];
    bet[i] = (const float*)d_in[base + 5];
    rm[i]  = (const float*)d_in[base + 6];
    rv[i]  = (const float*)d_in[base + 7];
  }
  const float* Wl = (const float*)d_in[28];
  const float* bl = (const float*)d_in[29];

  const int* src = edge_index;
  const int* dst = edge_index + N_EDGES;

  // Workspace layout (floats): agg | hA | hB | partial | hpool  (~312 MB)
  float* ws      = (float*)d_ws;
  float* agg     = ws;
  float* hA      = agg + (size_t)N_NODES * HDIM;
  float* hB      = hA  + (size_t)N_NODES * HDIM;
  float* partial = hB  + (size_t)N_NODES * HDIM;
  float* hpool   = partial + N_NODES;

  const int edge_blocks = (N_EDGES / 16 + 7) / 8;   // 8 waves/block
  const int node_blocks = (N_NODES / 16 + 7) / 8;
  const int nthr_blocks = (N_NODES + 255) / 256;

  // ---- Layer 1 (D = 64) ----
  zero_kernel<<<2048, 256, 0, stream>>>(agg, N_NODES * IN_DIM);
  edge_msg_kernel<IN_DIM><<<edge_blocks, 256, 0, stream>>>(
      x, src, dst, edge_attr, We[0], be[0], agg);
  node_gemm_kernel<IN_DIM><<<node_blocks, 256, 0, stream>>>(
      x, agg, W[0], b[0], gam[0], bet[0], rm[0], rv[0], hA);
  dot_accum_kernel<<<nthr_blocks, 256, 0, stream>>>(hA, Wl + 0 * HDIM, partial, 0);

  // ---- Layer 2 (D = 128) ----
  zero_kernel<<<2048, 256, 0, stream>>>(agg, N_NODES * HDIM);
  edge_msg_kernel<HDIM><<<edge_blocks, 256, 0, stream>>>(
      hA, src, dst, edge_attr, We[1], be[1], agg);
  node_gemm_kernel<HDIM><<<node_blocks, 256, 0, stream>>>(
      hA, agg, W[1], b[1], gam[1], bet[1], rm[1], rv[1], hB);
  dot_accum_kernel<<<nthr_blocks, 256, 0, stream>>>(hB, Wl + 1 * HDIM, partial, 1);

  // ---- Layer 3 (D = 128), h3 overwrites hA (h1's head term already banked) ----
  zero_kernel<<<2048, 256, 0, stream>>>(agg, N_NODES * HDIM);
  edge_msg_kernel<HDIM><<<edge_blocks, 256, 0, stream>>>(
      hB, src, dst, edge_attr, We[2], be[2], agg);
  node_gemm_kernel<HDIM><<<node_blocks, 256, 0, stream>>>(
      hB, agg, W[2], b[2], gam[2], bet[2], rm[2], rv[2], hA);
  dot_accum_kernel<<<nthr_blocks, 256, 0, stream>>>(hA, Wl + 2 * HDIM, partial, 1);

  // ---- Pool + head ----
  zero_kernel<<<256, 256, 0, stream>>>(hpool, N_MOLS * HDIM);
  pool_kernel<<<(N_NODES * HDIM + 255) / 256, 256, 0, stream>>>(hA, batch, hpool);
  final_kernel<<<nthr_blocks, 256, 0, stream>>>(
      partial, batch, hpool, Wl + 3 * HDIM, bl, (float*)d_out);
}